// UHGConv_65438121721900
// MI455X (gfx1250) — compile-verified
//
#include <hip/hip_runtime.h>

typedef __attribute__((ext_vector_type(2))) float v2f;
typedef __attribute__((ext_vector_type(8))) float v8f;

#define C_DIM 128
#define EPS_NORM 1e-8f
#define EPS_ACOSH 1e-7f

// ---------------------------------------------------------------------------
// Phase 1: one wave32 per edge. Lane l owns channels [4l, 4l+4).
// Minkowski inner product (last coord time-like), weight = exp(-acosh(m))
//   = 1/(m + sqrt(m*m - 1))  -- no transcendental needed.
// Scatter-add via native f32 global atomics (all traffic is L2-resident:
// x (25.6MB) + summed (25.6MB) << 192MB L2).
// ---------------------------------------------------------------------------
__global__ void uhg_edge_kernel(const float* __restrict__ x,
                                const int* __restrict__ ei,
                                float* __restrict__ summed,
                                unsigned* __restrict__ counts,
                                int E) {
  const int lane = threadIdx.x & 31;
  const int wave = (blockIdx.x * blockDim.x + threadIdx.x) >> 5;
  const int nwaves = (gridDim.x * blockDim.x) >> 5;
  const float sgn = (lane == 31) ? -1.0f : 1.0f;  // channel 127 is time-like

  for (int e = wave; e < E; e += nwaves) {
    const int row = ei[e];
    const int col = ei[E + e];
    const float4 xi = *(const float4*)(x + (size_t)row * C_DIM + lane * 4);
    const float4 xj = *(const float4*)(x + (size_t)col * C_DIM + lane * 4);

    float p = xi.x * xj.x + xi.y * xj.y + xi.z * xj.z + sgn * (xi.w * xj.w);
#pragma unroll
    for (int o = 16; o > 0; o >>= 1) p += __shfl_xor(p, o, 32);

    const float m = fmaxf(-p, 1.0f + EPS_ACOSH);
    const float w = 1.0f / (m + sqrtf(m * m - 1.0f));  // exp(-acosh(m))

    float* dst = summed + (size_t)row * C_DIM + lane * 4;
    unsafeAtomicAdd(dst + 0, xj.x * w);
    unsafeAtomicAdd(dst + 1, xj.y * w);
    unsafeAtomicAdd(dst + 2, xj.z * w);
    unsafeAtomicAdd(dst + 3, xj.w * w);
    if (lane == 0) atomicAdd(&counts[row], 1u);
  }
}

// ---------------------------------------------------------------------------
// Phase 2: out = (summed @ W^T) * inv_count[row] + bias ; L2 row-normalize.
// 3125 blocks x 256 threads; block owns a 16-row slab (50000 = 3125*16).
// Wave w computes the 16x16 tile at columns [16w,16w+16) with f32 WMMA
// (V_WMMA_F32_16X16X4_F32, K=128 -> 32 MACs, dual accumulator chains).
// Count division is deferred to the accumulator epilogue (it commutes with
// the GEMM), so nothing sits between the global loads and the WMMAs.
// Tile staged to LDS, then block-cooperative row normalization (fused).
// ---------------------------------------------------------------------------
__global__ void uhg_gemm_norm_kernel(const float* __restrict__ summed,
                                     const unsigned* __restrict__ counts,
                                     const float* __restrict__ weight,
                                     const float* __restrict__ bias,
                                     float* __restrict__ out,
                                     int Nn) {
  __shared__ float lds[16 * 132];  // stride 132 to avoid bank conflicts

  const int m0   = blockIdx.x * 16;
  const int wave = threadIdx.x >> 5;
  const int lane = threadIdx.x & 31;
  const int half = lane >> 4;      // K sub-group for A/B fragments
  const int l15  = lane & 15;
  const int n0   = wave * 16;

  // A fragment: row = m0+l15, K = 4*kk + 2*half + {0,1}
  int arow = m0 + l15;
  if (arow >= Nn) arow = Nn - 1;
  const float* aptr = summed + (size_t)arow * C_DIM + half * 2;
  // B fragment: B[k][n] = weight[n][k], n = n0+l15
  const float* bptr = weight + (size_t)(n0 + l15) * C_DIM + half * 2;

  // One reciprocal per lane: lane l (l15 = l&15) covers slab row m0+l15.
  const float inv_cnt_lane = 1.0f / fmaxf((float)counts[arow], 1.0f);

  v8f acc0 = {};
  v8f acc1 = {};
#pragma unroll
  for (int kk = 0; kk < 32; kk += 2) {
    v2f a0 = *(const v2f*)(aptr + 4 * kk);
    v2f b0 = *(const v2f*)(bptr + 4 * kk);
    acc0 = __builtin_amdgcn_wmma_f32_16x16x4_f32(false, a0, false, b0,
                                                 (short)0, acc0, false, false);
    v2f a1 = *(const v2f*)(aptr + 4 * kk + 4);
    v2f b1 = *(const v2f*)(bptr + 4 * kk + 4);
    acc1 = __builtin_amdgcn_wmma_f32_16x16x4_f32(false, a1, false, b1,
                                                 (short)0, acc1, false, false);
  }

  v8f acc = acc0 + acc1;
  const float bn = bias[n0 + l15];

  // C/D layout: VGPR r, lanes<16 -> M=r, lanes>=16 -> M=r+8; N = l15.
  // Scale row r's partial sums by inv_cnt of row (r + half*8), add bias.
#pragma unroll
  for (int r = 0; r < 8; ++r) {
    const float invc = __shfl(inv_cnt_lane, r + half * 8, 32);
    acc[r] = acc[r] * invc + bn;
  }

#pragma unroll
  for (int r = 0; r < 8; ++r)
    lds[(r + half * 8) * 132 + n0 + l15] = acc[r];

  __syncthreads();

  // Normalize: wave w handles rows 2w (lanes 0-15) and 2w+1 (lanes 16-31);
  // lane owns 8 consecutive columns.
  const int rloc = wave * 2 + half;
  const int grow = m0 + rloc;
  float v[8];
  float s = 0.0f;
#pragma unroll
  for (int j = 0; j < 8; ++j) {
    v[j] = lds[rloc * 132 + l15 * 8 + j];
    s += v[j] * v[j];
  }
#pragma unroll
  for (int o = 8; o > 0; o >>= 1) s += __shfl_xor(s, o, 32);
  const float scale = 1.0f / (sqrtf(s) + EPS_NORM);

  if (grow < Nn) {
    float* optr = out + (size_t)grow * C_DIM + l15 * 8;
#pragma unroll
    for (int j = 0; j < 8; ++j) optr[j] = v[j] * scale;
  }
}

// ---------------------------------------------------------------------------
extern "C" void kernel_launch(void* const* d_in, const int* in_sizes, int n_in,
                              void* d_out, int out_size, void* d_ws, size_t ws_size,
                              hipStream_t stream) {
  const float* x      = (const float*)d_in[0];
  const int*   ei     = (const int*)d_in[1];
  const float* weight = (const float*)d_in[2];
  const float* bias   = (const float*)d_in[3];
  float*       out    = (float*)d_out;

  const int Nn = in_sizes[0] / C_DIM;   // 50000
  const int E  = in_sizes[1] / 2;       // 600000

  float*    summed = (float*)d_ws;
  unsigned* counts = (unsigned*)((char*)d_ws + (size_t)Nn * C_DIM * sizeof(float));
  const size_t zbytes = (size_t)Nn * C_DIM * sizeof(float) + (size_t)Nn * sizeof(unsigned);
  hipMemsetAsync(d_ws, 0, zbytes, stream);

  uhg_edge_kernel<<<1024, 256, 0, stream>>>(x, ei, summed, counts, E);

  const int mtiles = (Nn + 15) / 16;    // 3125
  uhg_gemm_norm_kernel<<<mtiles, 256, 0, stream>>>(summed, counts, weight, bias, out, Nn);
}